// GraphProp_40261023433324
// MI455X (gfx1250) — compile-verified
//
#include <hip/hip_runtime.h>
#include <stdint.h>

#define NTOT 50000
#define LVLS 16
#define PER  3125   // NTOT / LVLS
#define DEG  16
#define DIM  64

// ---------------------------------------------------------------------------
// Kernel 1: copy level-0 feat rows into the output state buffer (the only
// rows never rewritten by the propagation), and warm the 192MB L2 with
// global_prefetch_b8 over delay / intra_delay (12.8 MB each; both stay
// L2-resident, and every later level gathers from them randomly).
// ---------------------------------------------------------------------------
__global__ void __launch_bounds__(256) warm_init_kernel(
    const float* __restrict__ feat,
    const float* __restrict__ delay,
    const float* __restrict__ intra,
    float*       __restrict__ out)
{
    const int tid = blockIdx.x * blockDim.x + threadIdx.x;

    // Copy level-0 rows: PER * DIM = 200000 floats.
    const int n0 = PER * DIM;
    if (tid < n0) out[tid] = feat[tid];

    // One 128-byte line prefetched per thread per array.
    const int nlines = (NTOT * DIM * 4) / 128;   // 100000
    if (tid < nlines) {
        __builtin_prefetch((const char*)delay + (size_t)tid * 128, 0, 1);
        __builtin_prefetch((const char*)intra + (size_t)tid * 128, 0, 1);
    }
}

// ---------------------------------------------------------------------------
// Kernel 2: one level of topological propagation.
//   feat[v] = max_k( feat[src_k] + intra[src_k] ) + delay[v]
// One wave32 per dst node, 2 floats per lane (b64 loads -> 256B contiguous
// per wave per request). All 16 edge indices are loaded up front (4 x int4,
// one wait), readfirstlane'd to SGPRs, and only then are the 32 row-gather
// loads issued so LOADcnt can pipeline them deeply. The per-node delay row
// is staged LDS-side with an async global->LDS load that overlaps the whole
// gather phase, synchronized with s_wait_asynccnt.
// ---------------------------------------------------------------------------
__global__ void __launch_bounds__(256) level_kernel(
    float*       out,                       // feat state, updated in place
    const float* __restrict__ intra,
    const float* __restrict__ delay,
    const int*   __restrict__ src,
    int level_start)
{
    __shared__ float2 s_delay[256];

    const int lane = threadIdx.x & 31;
    const int v    = level_start + (int)blockIdx.x * 8 + (threadIdx.x >> 5);
    if (v >= level_start + PER) return;     // wave-uniform exit

    // Async-stage this node's delay pair (8B/lane) into LDS; overlaps the
    // gather phase below. Tracked by ASYNCcnt, not LOADcnt.
    {
        uint32_t lds_addr = (uint32_t)(uintptr_t)&s_delay[threadIdx.x];
        const float2* gptr = (const float2*)delay + (size_t)v * (DIM / 2) + lane;
        asm volatile("global_load_async_to_lds_b64 %0, %1, off"
                     :: "v"(lds_addr), "v"(gptr)
                     : "memory");
    }

    // Batch-load all 16 edge indices (64B, wave-uniform address): 4 x b128,
    // a single loadcnt wait, then scalarize. This removes the per-edge
    // load->wait->readfirstlane->gather serialization.
    const int ebase = (v - PER) * DEG;
    const int4* sp = (const int4*)(src + ebase);
    int4 q0 = sp[0];
    int4 q1 = sp[1];
    int4 q2 = sp[2];
    int4 q3 = sp[3];
    int sidx[DEG];
    sidx[0]  = __builtin_amdgcn_readfirstlane(q0.x);
    sidx[1]  = __builtin_amdgcn_readfirstlane(q0.y);
    sidx[2]  = __builtin_amdgcn_readfirstlane(q0.z);
    sidx[3]  = __builtin_amdgcn_readfirstlane(q0.w);
    sidx[4]  = __builtin_amdgcn_readfirstlane(q1.x);
    sidx[5]  = __builtin_amdgcn_readfirstlane(q1.y);
    sidx[6]  = __builtin_amdgcn_readfirstlane(q1.z);
    sidx[7]  = __builtin_amdgcn_readfirstlane(q1.w);
    sidx[8]  = __builtin_amdgcn_readfirstlane(q2.x);
    sidx[9]  = __builtin_amdgcn_readfirstlane(q2.y);
    sidx[10] = __builtin_amdgcn_readfirstlane(q2.z);
    sidx[11] = __builtin_amdgcn_readfirstlane(q2.w);
    sidx[12] = __builtin_amdgcn_readfirstlane(q3.x);
    sidx[13] = __builtin_amdgcn_readfirstlane(q3.y);
    sidx[14] = __builtin_amdgcn_readfirstlane(q3.z);
    sidx[15] = __builtin_amdgcn_readfirstlane(q3.w);

    const float2* feat2  = (const float2*)out;     // rows < level_start: final
    const float2* intra2 = (const float2*)intra;

    // Two independent accumulator chains for a little extra ILP.
    float ax0 = -__builtin_huge_valf(), ay0 = -__builtin_huge_valf();
    float ax1 = -__builtin_huge_valf(), ay1 = -__builtin_huge_valf();

#pragma unroll
    for (int k = 0; k < DEG; k += 2) {
        size_t off0 = (size_t)sidx[k]     * (DIM / 2) + lane;
        size_t off1 = (size_t)sidx[k + 1] * (DIM / 2) + lane;
        float2 f0 = feat2[off0];
        float2 i0 = intra2[off0];
        float2 f1 = feat2[off1];
        float2 i1 = intra2[off1];
        ax0 = fmaxf(ax0, f0.x + i0.x);
        ay0 = fmaxf(ay0, f0.y + i0.y);
        ax1 = fmaxf(ax1, f1.x + i1.x);
        ay1 = fmaxf(ay1, f1.y + i1.y);
    }
    float ax = fmaxf(ax0, ax1);
    float ay = fmaxf(ay0, ay1);

    // Wait for the async LDS fill (this wave's own ASYNCcnt), then consume.
    asm volatile("s_wait_asynccnt 0x0" ::: "memory");
    float2 dl = s_delay[threadIdx.x];

    float2 r;
    r.x = ax + dl.x;
    r.y = ay + dl.y;
    ((float2*)out)[(size_t)v * (DIM / 2) + lane] = r;
}

// ---------------------------------------------------------------------------
// Launcher: init/warm kernel, then 15 sequential level kernels (stream order
// provides the topological dependency). All state lives in d_out.
// ---------------------------------------------------------------------------
extern "C" void kernel_launch(void* const* d_in, const int* in_sizes, int n_in,
                              void* d_out, int out_size, void* d_ws, size_t ws_size,
                              hipStream_t stream) {
    const float* feat  = (const float*)d_in[0];
    const float* delay = (const float*)d_in[1];
    const float* intra = (const float*)d_in[2];
    const int*   src   = (const int*)d_in[3];
    float* out = (float*)d_out;
    (void)in_sizes; (void)n_in; (void)out_size; (void)d_ws; (void)ws_size;

    {
        const int threads = PER * DIM;              // 200000 (>= prefetch lines)
        const int blocks  = (threads + 255) / 256;
        warm_init_kernel<<<blocks, 256, 0, stream>>>(feat, delay, intra, out);
    }

    for (int l = 1; l < LVLS; ++l) {
        const int blocks = (PER + 7) / 8;           // 8 nodes per 256-thread block
        level_kernel<<<blocks, 256, 0, stream>>>(out, intra, delay, src, l * PER);
    }
}